// Additive_63874753626387
// MI455X (gfx1250) — compile-verified
//
#include <hip/hip_runtime.h>
#include <math.h>

// ---------------------------------------------------------------------------
// Additive attention scores on gfx1250 (MI455X, wave32, WMMA).
//   Phase 1: qU = Q@U + b, kT = K@T        -> V_WMMA_F32_16X16X4_F32 (fp32)
//   Phase 2: out[b,q,k] = sum_h tanh(qU[b,q,h]+kT[b,k,h]) * w[h] + w_bias
//            -> LDS-tiled VALU + V_TANH_F32 (hardware transcendental)
// ---------------------------------------------------------------------------

typedef __attribute__((ext_vector_type(2))) float v2f;
typedef __attribute__((ext_vector_type(8))) float v8f;

#if __has_builtin(__builtin_amdgcn_tanhf)
  #define TANHF(x) __builtin_amdgcn_tanhf(x)
#elif __has_builtin(__builtin_amdgcn_tanh_f32)
  #define TANHF(x) __builtin_amdgcn_tanh_f32(x)
#else
  #define TANHF(x) tanhf(x)
#endif

#define HD    128          // hidden dim H
#define LSEQ  512          // Lq == Lk
#define NB    4            // batch
#define ROWS  (NB * LSEQ)  // 2048 rows per projection

// ===========================================================================
// Kernel 1: fp32 WMMA projections. One wave (32 threads) per 16x16 D-tile.
// tiles 0..1023  -> qU = queries @ U + b
// tiles 1024..2047 -> kT = keys    @ T
//
// A-frag layout (16x4 f32): lanes 0-15 hold M=0..15;
//   v[0] = A[M][k0 + 2*half], v[1] = A[M][k0 + 2*half + 1]
// B-frag layout (4x16 f32, mirrored): lanes 0-15 hold N=0..15;
//   v[0] = B[k0 + 2*half][N], v[1] = B[k0 + 2*half + 1][N]
// D layout: vgpr r, half h -> D[m0 + 8*h + r][n0 + l16]
// ===========================================================================
__global__ __launch_bounds__(32)
void proj_wmma_kernel(const float* __restrict__ q,
                      const float* __restrict__ k,
                      const float* __restrict__ U,
                      const float* __restrict__ T,
                      const float* __restrict__ bias,
                      float* __restrict__ qU,
                      float* __restrict__ kT)
{
    const int tile = blockIdx.x;                 // 0..2047
    const bool isQ = tile < 1024;
    const int  t   = isQ ? tile : (tile - 1024);
    const int  m0  = (t & 127) << 4;             // 128 M-tiles (rows)
    const int  n0  = (t >> 7)  << 4;             // 8 N-tiles (cols)

    const float* __restrict__ A  = isQ ? q : k;  // (2048 x 128)
    const float* __restrict__ Bm = isQ ? U : T;  // (128 x 128) row-major
    float*       __restrict__ D  = isQ ? qU : kT;

    const int lane = threadIdx.x & 31;
    const int half = lane >> 4;                  // 0 or 1
    const int l16  = lane & 15;

    const float* arow = A + (m0 + l16) * HD;

    v8f acc = {0.f, 0.f, 0.f, 0.f, 0.f, 0.f, 0.f, 0.f};

    #pragma unroll
    for (int kk = 0; kk < HD; kk += 4) {
        const int ka = kk + 2 * half;            // even -> float2-aligned
        const float2 a2 = *(const float2*)(arow + ka);
        v2f af; af.x = a2.x; af.y = a2.y;
        v2f bf;
        bf.x = Bm[(ka + 0) * HD + n0 + l16];
        bf.y = Bm[(ka + 1) * HD + n0 + l16];
        acc = __builtin_amdgcn_wmma_f32_16x16x4_f32(
                  /*neg_a=*/false, af, /*neg_b=*/false, bf,
                  /*c_mod=*/(short)0, acc, /*reuse_a=*/false, /*reuse_b=*/false);
    }

    const float bv = isQ ? bias[n0 + l16] : 0.0f;   // fold b into qU only
    #pragma unroll
    for (int r = 0; r < 8; ++r) {
        D[(m0 + 8 * half + r) * HD + n0 + l16] = acc[r] + bv;
    }
}

// ===========================================================================
// Kernel 2: tiled tanh-reduce.
//   grid: NB * (512/64) * (512/32) = 512 blocks, 256 threads (8 waves)
//   block tile: 64 q-rows x 32 k-rows; thread micro-tile: 4q x 2k
//   LDS: qs 64x132f + ks 32x132f + w  (~50 KB, mult. WGs per 320KB WGP)
//   stride 132: float4-aligned (528B) and k-row reads hit distinct banks
// ===========================================================================
#define TQ   64
#define TK   32
#define ROWP 132

__global__ __launch_bounds__(256)
void score_kernel(const float* __restrict__ qU,
                  const float* __restrict__ kT,
                  const float* __restrict__ w,
                  const float* __restrict__ wb,
                  float* __restrict__ out)
{
    __shared__ float qs[TQ * ROWP];
    __shared__ float ks[TK * ROWP];
    __shared__ float wl[HD];

    const int blk = blockIdx.x;          // 0..511
    const int b   = blk >> 7;            // /128
    const int ti  = blk & 127;
    const int q0  = (ti >> 4) * TQ;      // 8 q-tiles
    const int k0  = (ti & 15) * TK;      // 16 k-tiles

    const int tid = threadIdx.x;

    if (tid < HD) wl[tid] = w[tid];

    // Stage tiles: q tile = 64*32 float4, k tile = 32*32 float4
    const float4* qg = (const float4*)(qU + (b * LSEQ + q0) * HD);
    const float4* kg = (const float4*)(kT + (b * LSEQ + k0) * HD);
    #pragma unroll
    for (int it = 0; it < 8; ++it) {
        const int f   = tid + it * 256;      // 0..2047
        const int row = f >> 5;              // /32 (32 float4 per 128-col row)
        const int c4  = f & 31;
        *(float4*)(&qs[row * ROWP + c4 * 4]) = qg[row * 32 + c4];
        if (f < TK * 32) {
            *(float4*)(&ks[row * ROWP + c4 * 4]) = kg[row * 32 + c4];
        }
    }
    __syncthreads();

    const int tx = tid & 15;             // k group: k = k0 + tx + 16*j
    const int ty = tid >> 4;             // q group: q = q0 + ty + 16*i

    float acc[4][2];
    #pragma unroll
    for (int i = 0; i < 4; ++i)
        #pragma unroll
        for (int j = 0; j < 2; ++j)
            acc[i][j] = 0.0f;

    #pragma unroll 4
    for (int h = 0; h < HD; ++h) {
        const float wv = wl[h];
        float qv[4], kv[2];
        #pragma unroll
        for (int i = 0; i < 4; ++i) qv[i] = qs[(ty + 16 * i) * ROWP + h];
        #pragma unroll
        for (int j = 0; j < 2; ++j) kv[j] = ks[(tx + 16 * j) * ROWP + h];
        #pragma unroll
        for (int i = 0; i < 4; ++i)
            #pragma unroll
            for (int j = 0; j < 2; ++j)
                acc[i][j] = fmaf(TANHF(qv[i] + kv[j]), wv, acc[i][j]);
    }

    const float wbias = wb[0];
    #pragma unroll
    for (int i = 0; i < 4; ++i) {
        const int qr = q0 + ty + 16 * i;
        #pragma unroll
        for (int j = 0; j < 2; ++j) {
            const int kc = k0 + tx + 16 * j;
            out[(b * LSEQ + qr) * (size_t)LSEQ + kc] = acc[i][j] + wbias;
        }
    }
}

// ===========================================================================
// Launch: inputs = queries, keys, U, T, b, w, w_bias (fp32)
//         out    = (B, Lq, Lk) fp32 ; ws needs 2 MB for qU/kT
// ===========================================================================
extern "C" void kernel_launch(void* const* d_in, const int* in_sizes, int n_in,
                              void* d_out, int out_size, void* d_ws, size_t ws_size,
                              hipStream_t stream)
{
    const float* q    = (const float*)d_in[0];
    const float* k    = (const float*)d_in[1];
    const float* U    = (const float*)d_in[2];
    const float* T    = (const float*)d_in[3];
    const float* bias = (const float*)d_in[4];
    const float* w    = (const float*)d_in[5];
    const float* wb   = (const float*)d_in[6];
    float* out = (float*)d_out;

    float* qU = (float*)d_ws;                 // 2048*128 f32 = 1 MB
    float* kT = qU + (size_t)ROWS * HD;       // +1 MB

    proj_wmma_kernel<<<2048, 32, 0, stream>>>(q, k, U, T, bias, qU, kT);
    score_kernel<<<NB * 8 * 16, 256, 0, stream>>>(qU, kT, w, wb, out);
}